// ChainGNN_25202868093554
// MI455X (gfx1250) — compile-verified
//
#include <hip/hip_runtime.h>
#include <math.h>

#define TPB 256

typedef __attribute__((ext_vector_type(2))) float v2f;
typedef __attribute__((ext_vector_type(8))) float v8f;

__device__ __forceinline__ float lrelu(float v) { return v > 0.0f ? v : 0.2f * v; }

// fp32 atomic max via signed/unsigned int trick (valid with -inf init).
__device__ __forceinline__ void atomicMaxF(float* a, float v) {
  if (v >= 0.0f) atomicMax((int*)a, __float_as_int(v));
  else           atomicMin((unsigned int*)a, __float_as_uint(v));
}

// ---------------------------------------------------------------------------
// GEMM C[M,Nc] = A[M,K] @ B[K,Nc], fp32, one wave per 16x16 tile using
// V_WMMA_F32_16X16X4_F32.  ISA 7.12.2 layouts:
//   A 16x4 : lanes 0-15 hold {K=k,k+1} of row M=lane; lanes 16-31 hold {k+2,k+3}
//   B 4x16 : VGPR v: lanes 0-15 = row K=k+v, lanes 16-31 = row K=k+2+v, col N=lane&15
//   C/D    : VGPR v: lanes 0-15 -> M=v, lanes 16-31 -> M=v+8, col N=lane&15
// Requires M%16==0 handled by guard (wave-uniform), K%4==0, Nc%16==0.
// ---------------------------------------------------------------------------
__global__ void gat_gemm_wmma(const float* __restrict__ A, const float* __restrict__ B,
                              float* __restrict__ C, int M, int K, int Nc) {
  const int lane = threadIdx.x;      // 0..31
  const int half = lane >> 4;        // 0 | 1
  const int l16  = lane & 15;
  const int col0 = blockIdx.x * 16;
  const int row0 = (blockIdx.y * blockDim.y + threadIdx.y) * 16;
  if (row0 >= M) return;             // wave-uniform: EXEC stays all-ones for WMMA
  const int koff = half * 2;

  v8f acc = {};
  const float* arow = A + (size_t)(row0 + l16) * K + koff;       // A[row][koff+..]
  const float* bcol = B + (size_t)koff * Nc + col0 + l16;        // B[koff][col]
  for (int k = 0; k < K; k += 4) {
    v2f a, b;
    a[0] = arow[k];
    a[1] = arow[k + 1];
    b[0] = bcol[(size_t)k * Nc];
    b[1] = bcol[(size_t)(k + 1) * Nc];
    acc = __builtin_amdgcn_wmma_f32_16x16x4_f32(false, a, false, b,
                                                (short)0, acc, false, false);
  }
  float* crow = C + (size_t)(row0 + half * 8) * Nc + col0 + l16;
#pragma unroll
  for (int v = 0; v < 8; ++v)
    crow[(size_t)v * Nc] = acc[v];
}

// ---------------------------------------------------------------------------

__global__ void fill_f32(float* __restrict__ p, float v, long long n) {
  long long i = (long long)blockIdx.x * blockDim.x + threadIdx.x;
  if (i < n) p[i] = v;
}

// per (node, head): alpha_s = h[n,h,:].a_src[h,:],  alpha_d likewise
__global__ void node_alpha(const float* __restrict__ h, const float* __restrict__ a_src,
                           const float* __restrict__ a_dst, float* __restrict__ as,
                           float* __restrict__ ad, int N, int heads, int ch) {
  int i = blockIdx.x * blockDim.x + threadIdx.x;
  if (i >= N * heads) return;
  int n = i / heads, hd = i - n * heads;
  const float* hp = h + (size_t)n * heads * ch + (size_t)hd * ch;
  const float* sw = a_src + hd * ch;
  const float* dw = a_dst + hd * ch;
  float s = 0.f, d = 0.f;
  for (int c = 0; c < ch; ++c) { float v = hp[c]; s += v * sw[c]; d += v * dw[c]; }
  as[i] = s; ad[i] = d;
}

// edge_index layout: ei[0..E-1]=src, ei[E..2E-1]=dst; edges E..E+N-1 are self loops
__global__ void edge_max(const int* __restrict__ ei, const float* __restrict__ as,
                         const float* __restrict__ ad, float* __restrict__ m,
                         int E, int N, int heads) {
  int e = blockIdx.x * blockDim.x + threadIdx.x;
  if (e >= E + N) return;
  int s = (e < E) ? ei[e] : (e - E);
  int d = (e < E) ? ei[E + e] : (e - E);
  for (int h = 0; h < heads; ++h) {
    float l = lrelu(as[s * heads + h] + ad[d * heads + h]);
    atomicMaxF(&m[d * heads + h], l);
  }
}

__global__ void edge_expsum(const int* __restrict__ ei, const float* __restrict__ as,
                            const float* __restrict__ ad, const float* __restrict__ m,
                            float* __restrict__ den, float* __restrict__ ee,
                            int E, int N, int heads) {
  int e = blockIdx.x * blockDim.x + threadIdx.x;
  if (e >= E + N) return;
  int s = (e < E) ? ei[e] : (e - E);
  int d = (e < E) ? ei[E + e] : (e - E);
  for (int h = 0; h < heads; ++h) {
    float l  = lrelu(as[s * heads + h] + ad[d * heads + h]);
    float ex = expf(l - m[d * heads + h]);
    ee[(size_t)e * heads + h] = ex;
    atomicAdd(&den[d * heads + h], ex);
  }
}

__global__ void edge_norm(const int* __restrict__ ei, const float* __restrict__ den,
                          float* __restrict__ ee, int E, int N, int heads) {
  long long i = (long long)blockIdx.x * blockDim.x + threadIdx.x;
  long long total = (long long)(E + N) * heads;
  if (i >= total) return;
  int e = (int)(i / heads);
  int h = (int)(i - (long long)e * heads);
  int d = (e < E) ? ei[E + e] : (e - E);
  ee[i] = ee[i] / (den[d * heads + h] + 1e-16f);
}

// per (edge, channel): out[dst, c] += alpha[e, c/ch] * h[src, c]
__global__ void edge_agg(const int* __restrict__ ei, const float* __restrict__ h,
                         const float* __restrict__ alpha, float* __restrict__ out,
                         int E, int N, int heads, int ch) {
  long long i = (long long)blockIdx.x * blockDim.x + threadIdx.x;
  const int HC = heads * ch;
  long long total = (long long)(E + N) * HC;
  if (i >= total) return;
  int e  = (int)(i / HC);
  int c  = (int)(i - (long long)e * HC);
  int hd = c / ch;
  int s = (e < E) ? ei[e] : (e - E);
  int d = (e < E) ? ei[E + e] : (e - E);
  atomicAdd(&out[(size_t)d * HC + c],
            alpha[(size_t)e * heads + hd] * h[(size_t)s * HC + c]);
}

__global__ void bias_act(float* __restrict__ out, const float* __restrict__ bias,
                         int N, int HC, int relu) {
  int i = blockIdx.x * blockDim.x + threadIdx.x;
  if (i >= N * HC) return;
  float v = out[i] + bias[i % HC];
  out[i] = relu ? fmaxf(v, 0.f) : v;
}

// ---------------------------------------------------------------------------

extern "C" void kernel_launch(void* const* d_in, const int* in_sizes, int n_in,
                              void* d_out, int out_size, void* d_ws, size_t ws_size,
                              hipStream_t stream) {
  const float* x   = (const float*)d_in[0];
  const int*   ei  = (const int*)d_in[1];
  const float* W1  = (const float*)d_in[2];
  const float* A1s = (const float*)d_in[3];
  const float* A1d = (const float*)d_in[4];
  const float* b1  = (const float*)d_in[5];
  const float* W2  = (const float*)d_in[6];
  const float* A2s = (const float*)d_in[7];
  const float* A2d = (const float*)d_in[8];
  const float* b2  = (const float*)d_in[9];
  const float* W3  = (const float*)d_in[10];
  const float* A3s = (const float*)d_in[11];
  const float* A3d = (const float*)d_in[12];
  const float* b3  = (const float*)d_in[13];

  const int N  = in_sizes[0] / 128;
  const int E  = in_sizes[1] / 2;
  const int nE = E + N;

  // workspace carve-up (floats)
  float* bufA = (float*)d_ws;                     // h buffer   [N,128] max
  float* bufB = bufA + (size_t)N * 128;           // out buffer [N,128] max
  float* asb  = bufB + (size_t)N * 128;           // [N,4]
  float* adb  = asb  + (size_t)N * 4;             // [N,4]
  float* mb   = adb  + (size_t)N * 4;             // [N,4]
  float* denb = mb   + (size_t)N * 4;             // [N,4]
  float* eeb  = denb + (size_t)N * 4;             // [nE,4]

  auto run_layer = [&](const float* in, int K, const float* W, const float* a_s,
                       const float* a_d, const float* bias, int heads, int ch,
                       float* hbuf, float* obuf, int do_relu) {
    const int HC = heads * ch;
    // 1) h = in @ W  (WMMA fp32)
    {
      dim3 blk(32, 4, 1);
      dim3 grd(HC / 16, (N / 16 + 3) / 4, 1);
      gat_gemm_wmma<<<grd, blk, 0, stream>>>(in, W, hbuf, N, K, HC);
    }
    // 2) per-node attention scores
    {
      int t = N * heads;
      node_alpha<<<(t + TPB - 1) / TPB, TPB, 0, stream>>>(hbuf, a_s, a_d, asb, adb, N, heads, ch);
    }
    // 3) init reductions + output accumulator
    {
      long long nh = (long long)N * heads;
      fill_f32<<<(unsigned)((nh + TPB - 1) / TPB), TPB, 0, stream>>>(mb, -INFINITY, nh);
      fill_f32<<<(unsigned)((nh + TPB - 1) / TPB), TPB, 0, stream>>>(denb, 0.f, nh);
      long long no = (long long)N * HC;
      fill_f32<<<(unsigned)((no + TPB - 1) / TPB), TPB, 0, stream>>>(obuf, 0.f, no);
    }
    // 4) segment softmax over edges
    {
      int blks = (nE + TPB - 1) / TPB;
      edge_max   <<<blks, TPB, 0, stream>>>(ei, asb, adb, mb, E, N, heads);
      edge_expsum<<<blks, TPB, 0, stream>>>(ei, asb, adb, mb, denb, eeb, E, N, heads);
      long long tn = (long long)nE * heads;
      edge_norm<<<(unsigned)((tn + TPB - 1) / TPB), TPB, 0, stream>>>(ei, denb, eeb, E, N, heads);
    }
    // 5) weighted aggregation
    {
      long long tot = (long long)nE * HC;
      edge_agg<<<(unsigned)((tot + TPB - 1) / TPB), TPB, 0, stream>>>(ei, hbuf, eeb, obuf, E, N, heads, ch);
    }
    // 6) bias (+ReLU)
    {
      int t = N * HC;
      bias_act<<<(t + TPB - 1) / TPB, TPB, 0, stream>>>(obuf, bias, N, HC, do_relu);
    }
  };

  // layer 1: 128 -> 4x32 concat (128), relu
  run_layer(x,    128, W1, A1s, A1d, b1, 4, 32, bufA, bufB, 1);
  // layer 2: 128 -> 2x32 concat (64), relu   (in=bufB, h=bufA, out overwrites bufB
  //   only after the GEMM has consumed it — stream-ordered)
  run_layer(bufB, 128, W2, A2s, A2d, b2, 2, 32, bufA, bufB, 1);
  // layer 3: 64 -> 1x16 mean (16), no relu, straight into d_out
  run_layer(bufB, 64,  W3, A3s, A3d, b3, 1, 16, bufA, (float*)d_out, 0);
}